// SAGENet_69793218560204
// MI455X (gfx1250) — compile-verified
//
#include <hip/hip_runtime.h>
#include <hip/hip_bf16.h>
#include <math.h>

#define NNODES 100000
#define NEDGES 1600000
#define IN_CH  128
#define EDGE_CH 16
#define CAT    144      // IN_CH/HID + EDGE_CH
#define HID    128
#define OUTC   4
#define L2EPS  1e-12f

typedef __attribute__((ext_vector_type(2))) float v2f;
typedef __attribute__((ext_vector_type(8))) float v8f;

// ---------------------------------------------------------------- zero scratch
__global__ void zero_kernel(float* __restrict__ p, int n) {
    int i = blockIdx.x * blockDim.x + threadIdx.x;
    if (i < n) p[i] = 0.0f;
}

// --------------------------------------------------- edge phase: one wave/edge
// gather feat[row] (128f, b128 per lane) + edge_attr (16f), L2-normalize the
// 144-float message via wave32 shuffle reduction, scatter-add into agg[col].
__global__ __launch_bounds__(256)
void edge_scatter_kernel(const float* __restrict__ feat,
                         const float* __restrict__ eattr,
                         const int*   __restrict__ rowi,
                         const int*   __restrict__ coli,
                         float* __restrict__ agg,
                         float* __restrict__ cnt) {
    int gtid = blockIdx.x * blockDim.x + threadIdx.x;
    int e    = gtid >> 5;        // wave index == edge index
    int lane = gtid & 31;
    if (e >= NEDGES) return;

    int r = rowi[e];
    int c = coli[e];

    const float4* fx = (const float4*)(feat + (size_t)r * IN_CH);
    float4 xv = fx[lane];                               // 32 lanes x 4 = 128
    float4 ev = make_float4(0.f, 0.f, 0.f, 0.f);
    if (lane < 4) {
        const float4* fe = (const float4*)(eattr + (size_t)e * EDGE_CH);
        ev = fe[lane];                                  // 4 lanes x 4 = 16
    }
    float ss = xv.x*xv.x + xv.y*xv.y + xv.z*xv.z + xv.w*xv.w
             + ev.x*ev.x + ev.y*ev.y + ev.z*ev.z + ev.w*ev.w;
    #pragma unroll
    for (int off = 16; off > 0; off >>= 1)
        ss += __shfl_xor(ss, off, 32);
    float inv = 1.0f / fmaxf(sqrtf(ss), L2EPS);

    float* dst = agg + (size_t)c * CAT;
    atomicAdd(dst + lane*4 + 0, xv.x * inv);
    atomicAdd(dst + lane*4 + 1, xv.y * inv);
    atomicAdd(dst + lane*4 + 2, xv.z * inv);
    atomicAdd(dst + lane*4 + 3, xv.w * inv);
    if (lane < 4) {
        atomicAdd(dst + IN_CH + lane*4 + 0, ev.x * inv);
        atomicAdd(dst + IN_CH + lane*4 + 1, ev.y * inv);
        atomicAdd(dst + IN_CH + lane*4 + 2, ev.z * inv);
        atomicAdd(dst + IN_CH + lane*4 + 3, ev.w * inv);
    }
    if (lane == 0) atomicAdd(cnt + c, 1.0f);
}

// ------------------------------------------- layer-1 node phase (WMMA f32 GEMM)
// block = 256 thr = 8 waves, 16 output rows per block, wave w owns cols [16w,16w+16)
// h = relu(l2norm(mean_agg @ W1 + b1))
__global__ __launch_bounds__(256)
void layer1_node_kernel(const float* __restrict__ agg,
                        const float* __restrict__ cnt,
                        const float* __restrict__ W1,
                        const float* __restrict__ b1,
                        float* __restrict__ h) {
    __shared__ float sA[16][CAT + 1];    // mean-normalized A tile 16x144
    __shared__ float sO[16][HID + 1];    // output tile 16x128
    __shared__ float sInv[16];
    __shared__ float sNrm[16];

    int tid = threadIdx.x;
    int m0  = blockIdx.x * 16;

    if (tid < 16) sInv[tid] = 1.0f / fmaxf(cnt[m0 + tid], 1.0f);
    __syncthreads();
    for (int idx = tid; idx < 16 * CAT; idx += 256) {
        int rr = idx / CAT, kk = idx % CAT;
        sA[rr][kk] = agg[(size_t)(m0 + rr) * CAT + kk] * sInv[rr];
    }
    __syncthreads();

    int wave = tid >> 5, lane = tid & 31;
    int colbase = wave * 16;
    int n    = lane & 15;                 // B/D column within tile
    int arow = lane & 15;                 // A row
    int ksel = (lane < 16) ? 0 : 2;       // lanes 0-15: K=0,1  lanes 16-31: K=2,3

    v8f acc = {};
    for (int k0 = 0; k0 < CAT; k0 += 4) {
        v2f a, b;
        a[0] = sA[arow][k0 + ksel];
        a[1] = sA[arow][k0 + ksel + 1];
        b[0] = W1[(size_t)(k0 + ksel)     * HID + colbase + n];
        b[1] = W1[(size_t)(k0 + ksel + 1) * HID + colbase + n];
        acc = __builtin_amdgcn_wmma_f32_16x16x4_f32(false, a, false, b,
                                                    (short)0, acc, false, false);
    }
    float bias  = b1[colbase + n];
    int   rbase = (lane < 16) ? 0 : 8;    // C/D: vgpr r -> row r / r+8
    #pragma unroll
    for (int rr = 0; rr < 8; ++rr)
        sO[rbase + rr][colbase + n] = acc[rr] + bias;
    __syncthreads();

    if (tid < 16) {
        float ss = 0.f;
        for (int cc = 0; cc < HID; ++cc) { float v = sO[tid][cc]; ss += v * v; }
        sNrm[tid] = 1.0f / fmaxf(sqrtf(ss), L2EPS);
    }
    __syncthreads();
    for (int idx = tid; idx < 16 * HID; idx += 256) {
        int rr = idx / HID, cc = idx % HID;
        float v = sO[rr][cc] * sNrm[rr];
        h[(size_t)(m0 + rr) * HID + cc] = fmaxf(v, 0.0f);  // relu after l2norm
    }
}

// ------------------------------------------- layer-2 node phase (WMMA f32 GEMM)
// block = 128 thr = 4 waves, each wave owns one 16-row tile; only 4 valid cols.
// out = sigmoid(l2norm(mean_agg @ W2 + b2))
__global__ __launch_bounds__(128)
void layer2_node_kernel(const float* __restrict__ agg,
                        const float* __restrict__ cnt,
                        const float* __restrict__ W2,
                        const float* __restrict__ b2,
                        float* __restrict__ out) {
    __shared__ float sO[4][16][17];

    int tid = threadIdx.x;
    int wave = tid >> 5, lane = tid & 31;
    const int ntiles = NNODES / 16;            // 6250
    int tile = blockIdx.x * 4 + wave;
    if (tile >= ntiles) tile = ntiles - 1;     // duplicate last tile: same data, deterministic
    int m0 = tile * 16;

    int arow = lane & 15;
    int n    = lane & 15;
    int ksel = (lane < 16) ? 0 : 2;

    float ic = 1.0f / fmaxf(cnt[m0 + arow], 1.0f);
    const float* ap = agg + (size_t)(m0 + arow) * CAT;

    v8f acc = {};
    for (int k0 = 0; k0 < CAT; k0 += 4) {
        v2f a, b;
        a[0] = ap[k0 + ksel]     * ic;
        a[1] = ap[k0 + ksel + 1] * ic;
        b[0] = (n < OUTC) ? W2[(size_t)(k0 + ksel)     * OUTC + n] : 0.0f;
        b[1] = (n < OUTC) ? W2[(size_t)(k0 + ksel + 1) * OUTC + n] : 0.0f;
        acc = __builtin_amdgcn_wmma_f32_16x16x4_f32(false, a, false, b,
                                                    (short)0, acc, false, false);
    }
    float bias  = (n < OUTC) ? b2[n] : 0.0f;
    int   rbase = (lane < 16) ? 0 : 8;
    #pragma unroll
    for (int rr = 0; rr < 8; ++rr)
        sO[wave][rbase + rr][n] = acc[rr] + bias;
    __syncthreads();

    if (lane < 16) {
        float ss = 0.f;
        #pragma unroll
        for (int cc = 0; cc < OUTC; ++cc) { float v = sO[wave][lane][cc]; ss += v * v; }
        float invn = 1.0f / fmaxf(sqrtf(ss), L2EPS);
        #pragma unroll
        for (int cc = 0; cc < OUTC; ++cc) {
            float v = sO[wave][lane][cc] * invn;
            out[(size_t)(m0 + lane) * OUTC + cc] = 1.0f / (1.0f + __expf(-v));
        }
    }
}

// -----------------------------------------------------------------------------
extern "C" void kernel_launch(void* const* d_in, const int* in_sizes, int n_in,
                              void* d_out, int out_size, void* d_ws, size_t ws_size,
                              hipStream_t stream) {
    (void)in_sizes; (void)n_in; (void)out_size; (void)ws_size;

    const float* x     = (const float*)d_in[0];
    const int*   ei    = (const int*)  d_in[1];   // [2, E] int32
    const float* eattr = (const float*)d_in[2];
    const float* W1    = (const float*)d_in[3];
    const float* b1    = (const float*)d_in[4];
    const float* W2    = (const float*)d_in[5];
    const float* b2    = (const float*)d_in[6];
    float*       outp  = (float*)d_out;

    const int* rowi = ei;
    const int* coli = ei + NEDGES;

    float* agg = (float*)d_ws;                       // N*144
    float* cnt = agg + (size_t)NNODES * CAT;         // N
    float* h   = cnt + NNODES;                       // N*128

    const int nz = NNODES * (CAT + 1);               // agg+cnt contiguous
    dim3 zb(256), zg((nz + 255) / 256);
    dim3 eb(256), eg((NEDGES * 32 + 255) / 256);     // one wave per edge
    dim3 n1b(256), n1g(NNODES / 16);                 // 6250 row tiles
    dim3 n2b(128), n2g((NNODES / 16 + 3) / 4);       // 4 tiles/block

    // ---- layer 1
    zero_kernel<<<zg, zb, 0, stream>>>(agg, nz);
    edge_scatter_kernel<<<eg, eb, 0, stream>>>(x, eattr, rowi, coli, agg, cnt);
    layer1_node_kernel<<<n1g, n1b, 0, stream>>>(agg, cnt, W1, b1, h);
    // ---- layer 2
    zero_kernel<<<zg, zb, 0, stream>>>(agg, nz);
    edge_scatter_kernel<<<eg, eb, 0, stream>>>(h, eattr, rowi, coli, agg, cnt);
    layer2_node_kernel<<<n2g, n2b, 0, stream>>>(agg, cnt, W2, b2, outp);
}